// LukeMicronAblation_84344567759287
// MI455X (gfx1250) — compile-verified
//
#include <hip/hip_runtime.h>
#include <hip/hip_bf16.h>

typedef __attribute__((ext_vector_type(2))) float v2f;
typedef __attribute__((ext_vector_type(8))) float v8f;

#define BATCH   16384
#define LCODES  32
#define DDIM    128

// ---------------------------------------------------------------------------
// Kernel 1: weight transpose with zero-padding.  Wt[k*Npad + n] = W[n*K + k]
// ---------------------------------------------------------------------------
__global__ void transpose_pad_kernel(const float* __restrict__ W,
                                     float* __restrict__ Wt,
                                     int Nrows, int K, int Npad) {
    int idx = blockIdx.x * blockDim.x + threadIdx.x;
    if (idx >= K * Npad) return;
    int k = idx / Npad;
    int n = idx - k * Npad;
    Wt[idx] = (n < Nrows) ? W[n * K + k] : 0.0f;
}

// ---------------------------------------------------------------------------
// Kernel 2: four EmbeddingBag sums.
// One block (128 threads) per batch row. Codes staged through LDS.
// xcur[b] = [sum diag_emb[diag_codes] | sum proc_emb[proc_codes]]   (256 f32)
// xprev[b] likewise for prev codes.
// ---------------------------------------------------------------------------
__global__ void bag_kernel(const int* __restrict__ diag_codes,
                           const int* __restrict__ proc_codes,
                           const int* __restrict__ pdiag_codes,
                           const int* __restrict__ pproc_codes,
                           const float* __restrict__ diag_emb,
                           const float* __restrict__ proc_emb,
                           float* __restrict__ xcur,
                           float* __restrict__ xprev) {
    __shared__ int sc[4 * LCODES];
    const int b = blockIdx.x;
    const int d = threadIdx.x;            // 0..127

    // cooperative code load: group g = d>>5 picks the code array
    {
        const int g = d >> 5;
        const int l = d & 31;
        const int* p = (g == 0) ? diag_codes
                     : (g == 1) ? proc_codes
                     : (g == 2) ? pdiag_codes
                                : pproc_codes;
        sc[d] = p[b * LCODES + l];
    }
    __syncthreads();

    float s0 = 0.f, s1 = 0.f, s2 = 0.f, s3 = 0.f;
#pragma unroll 4
    for (int l = 0; l < LCODES; ++l) {
        s0 += diag_emb[(long)sc[l]               * DDIM + d];
        s1 += proc_emb[(long)sc[LCODES + l]      * DDIM + d];
        s2 += diag_emb[(long)sc[2 * LCODES + l]  * DDIM + d];
        s3 += proc_emb[(long)sc[3 * LCODES + l]  * DDIM + d];
    }
    float* xc = xcur  + (long)b * 256;
    float* xp = xprev + (long)b * 256;
    xc[d]        = s0;
    xc[DDIM + d] = s1;
    xp[d]        = s2;
    xp[DDIM + d] = s3;
}

// ---------------------------------------------------------------------------
// Kernel 3: generic fp32 WMMA GEMM with fused bias + activation.
//   Y[m, n] = act( sum_k X[m,k] * Wt[k,n] + bias[n] )
// One wave computes one 16x16 output tile via V_WMMA_F32_16X16X4_F32.
// 8 waves / block.  Tiles are linearized gid = mTile*nTiles + nTile.
// act: 0 = identity, 1 = relu, 2 = sigmoid.
// nMax guards stores/bias for padded N (W3: 1008 padded, 1000 real).
// ---------------------------------------------------------------------------
__global__ void gemm_wmma_kernel(const float* __restrict__ X, int ldX,
                                 const float* __restrict__ Wt, int ldW,
                                 const float* __restrict__ bias,
                                 float* __restrict__ Y, int ldY,
                                 int K, int nTiles, int nMax, int act) {
    const int wave = threadIdx.x >> 5;                 // 0..7 (wave32)
    const int lane = threadIdx.x & 31;
    const int half = lane >> 4;                        // 0 or 1
    const int lm   = lane & 15;

    const int gid   = blockIdx.x * 8 + wave;
    const int nT    = gid % nTiles;
    const int mT    = gid / nTiles;
    const int m0    = mT * 16;
    const int n0    = nT * 16;
    const int col   = n0 + lm;

    const float* Arow = X + (long)(m0 + lm) * ldX + 2 * half;
    const float* Bcol = Wt + (long)(2 * half) * ldW + col;

    v8f acc = {};
#pragma unroll 8
    for (int kk = 0; kk < K; kk += 4) {
        // A: element j holds X[m0+lm][kk + 2*half + j]  (contiguous pair)
        v2f a = *(const v2f*)(Arow + kk);
        // B: element j holds Wt[kk + 2*half + j][col]
        v2f b;
        b.x = Bcol[(long)kk * ldW];
        b.y = Bcol[(long)(kk + 1) * ldW];
        acc = __builtin_amdgcn_wmma_f32_16x16x4_f32(
            /*neg_a=*/false, a, /*neg_b=*/false, b,
            /*c_mod=*/(short)0, acc, /*reuse_a=*/false, /*reuse_b=*/false);
    }

    const bool valid = (col < nMax);
    const float bn = valid ? bias[col] : 0.0f;

#pragma unroll
    for (int j = 0; j < 8; ++j) {
        if (valid) {
            const int m = m0 + j + 8 * half;
            float y = acc[j] + bn;
            if (act == 1) y = fmaxf(y, 0.0f);
            else if (act == 2) y = 1.0f / (1.0f + __expf(-y));
            Y[(long)m * ldY + col] = y;
        }
    }
}

// ---------------------------------------------------------------------------
// Host-side launcher
// ---------------------------------------------------------------------------
extern "C" void kernel_launch(void* const* d_in, const int* in_sizes, int n_in,
                              void* d_out, int out_size, void* d_ws, size_t ws_size,
                              hipStream_t stream) {
    const int*   diag_codes = (const int*)  d_in[0];
    const int*   proc_codes = (const int*)  d_in[1];
    const int*   pdiag      = (const int*)  d_in[2];
    const int*   pproc      = (const int*)  d_in[3];
    const float* diag_emb   = (const float*)d_in[4];
    const float* proc_emb   = (const float*)d_in[5];
    const float* W1         = (const float*)d_in[6];   // [128, 256]
    const float* b1         = (const float*)d_in[7];   // [128]
    const float* W2         = (const float*)d_in[8];   // [256, 256]
    const float* b2         = (const float*)d_in[9];   // [256]
    const float* W3         = (const float*)d_in[10];  // [1000, 256]
    const float* b3         = (const float*)d_in[11];  // [1000]
    float*       out        = (float*)d_out;           // [B, 1000]

    // workspace layout (floats)
    float* ws   = (float*)d_ws;
    float* W1t  = ws;                        // 256 x 128
    float* W2t  = W1t + 256 * 128;           // 256 x 256
    float* W3t  = W2t + 256 * 256;           // 256 x 1008 (padded)
    float* xcur = W3t + 256 * 1008;          // B x 256
    float* xprev= xcur + (long)BATCH * 256;  // B x 256
    float* rep  = xprev + (long)BATCH * 256; // B x 256  = [cur | prev]
    float* h    = rep  + (long)BATCH * 256;  // B x 256

    // 1) weight transposes (tiny)
    transpose_pad_kernel<<<(256 * 128 + 255) / 256, 256, 0, stream>>>(W1, W1t, 128, 256, 128);
    transpose_pad_kernel<<<(256 * 256 + 255) / 256, 256, 0, stream>>>(W2, W2t, 256, 256, 256);
    transpose_pad_kernel<<<(256 * 1008 + 255) / 256, 256, 0, stream>>>(W3, W3t, 1000, 256, 1008);

    // 2) embedding bag sums
    bag_kernel<<<BATCH, DDIM, 0, stream>>>(diag_codes, proc_codes, pdiag, pproc,
                                           diag_emb, proc_emb, xcur, xprev);

    // 3) GEMM1: cur  = xcur  @ W1t + b1 -> rep[:, 0:128]
    //          prev = xprev @ W1t + b1 -> rep[:, 128:256]
    {
        const int nTiles = 128 / 16;                       // 8
        const int blocks = (BATCH / 16) * nTiles / 8;      // 1024
        gemm_wmma_kernel<<<blocks, 256, 0, stream>>>(xcur, 256, W1t, 128, b1,
                                                     rep, 256, 256, nTiles, 128, 0);
        gemm_wmma_kernel<<<blocks, 256, 0, stream>>>(xprev, 256, W1t, 128, b1,
                                                     rep + 128, 256, 256, nTiles, 128, 0);
    }

    // 4) GEMM2: h = relu(rep @ W2t + b2)
    {
        const int nTiles = 256 / 16;                       // 16
        const int blocks = (BATCH / 16) * nTiles / 8;      // 2048
        gemm_wmma_kernel<<<blocks, 256, 0, stream>>>(rep, 256, W2t, 256, b2,
                                                     h, 256, 256, nTiles, 256, 1);
    }

    // 5) GEMM3: out = sigmoid(h @ W3t + b3)   (N padded 1008 -> guard 1000)
    {
        const int nTiles = 1008 / 16;                      // 63
        const int blocks = (BATCH / 16) * nTiles / 8;      // 8064
        gemm_wmma_kernel<<<blocks, 256, 0, stream>>>(h, 256, W3t, 1008, b3,
                                                     out, 1000, 256, nTiles, 1000, 2);
    }
}